// LuongAttention_81312320847935
// MI455X (gfx1250) — compile-verified
//
#include <hip/hip_runtime.h>
#include <math.h>

typedef __attribute__((ext_vector_type(2))) float v2f;
typedef __attribute__((ext_vector_type(8))) float v8f;

#define Bn 32
#define Sn 4096
#define Hn 1024
#define SCH 8               // S-chunks for context partial sums
#define SLEN (Sn / SCH)     // 512 s-values per chunk

// ---------------------------------------------------------------------------
// Y[M,N] = X[M,K] @ W[N,K]^T + bias[N]   (fp32, WMMA 16x16x4)
// One wave (32 lanes) per 16x16 output tile; blockDim 256 = 8 waves/8 N-tiles.
// A-tile (ISA 7.12.2, 32-bit A 16x4): lane l<16 -> row m0+l, K=k..k+1 in
// v0,v1 ; lane l>=16 -> row m0+(l-16), K=k+2..k+3. B-tile of W^T uses the
// identical addressing on W rows (X.W^T is symmetric in X and W).
// C/D: VGPR r -> row m0 + r + 8*(lane>=16), col n0 + (lane&15).
// ---------------------------------------------------------------------------
__global__ __launch_bounds__(256)
void gemm_xwT_bias(const float* __restrict__ X, const float* __restrict__ W,
                   const float* __restrict__ bias, float* __restrict__ Y, int K)
{
    const int lane = threadIdx.x & 31;
    const int wid  = threadIdx.x >> 5;
    const int half = lane >> 4;
    const int l    = lane & 15;
    const int n0   = (blockIdx.x * 8 + wid) * 16;
    const int m0   = blockIdx.y * 16;

    v8f acc = {0.f, 0.f, 0.f, 0.f, 0.f, 0.f, 0.f, 0.f};
    const float* ap = X + (size_t)(m0 + l) * K + half * 2;
    const float* bp = W + (size_t)(n0 + l) * K + half * 2;

    for (int k = 0; k < K; k += 4) {
        v2f a, b;
        a.x = ap[k]; a.y = ap[k + 1];
        b.x = bp[k]; b.y = bp[k + 1];
        acc = __builtin_amdgcn_wmma_f32_16x16x4_f32(false, a, false, b,
                                                    (short)0, acc, false, false);
    }
    const float bv = bias[n0 + l];
#pragma unroll
    for (int r = 0; r < 8; ++r)
        Y[(size_t)(m0 + half * 8 + r) * Hn + n0 + l] = acc[r] + bv;
}

// ---------------------------------------------------------------------------
// scores[row] = dot(enc[row,:], h2[row/S,:])   row in [0, B*S)
// One wave per row; b128 loads, 512B contiguous per wave per iteration.
// ---------------------------------------------------------------------------
__global__ __launch_bounds__(256)
void scores_kernel(const float* __restrict__ enc, const float* __restrict__ h2,
                   float* __restrict__ scores)
{
    const int lane = threadIdx.x & 31;
    const int wid  = threadIdx.x >> 5;
    const long row = (long)blockIdx.x * 8 + wid;
    const int  b   = (int)(row >> 12);           // row / 4096

    const float4* e4 = reinterpret_cast<const float4*>(enc + row * (size_t)Hn);
    const float4* h4 = reinterpret_cast<const float4*>(h2 + (size_t)b * Hn);

    float acc = 0.f;
#pragma unroll
    for (int j = 0; j < 8; ++j) {
        const int i = lane + j * 32;             // 32 lanes * 8 iters * 4 = 1024
        float4 e = e4[i], h = h4[i];
        acc += e.x * h.x + e.y * h.y + e.z * h.z + e.w * h.w;
    }
#pragma unroll
    for (int off = 16; off > 0; off >>= 1)
        acc += __shfl_down(acc, off, 32);
    if (lane == 0) scores[row] = acc;
}

// ---------------------------------------------------------------------------
// attn[b,:] = softmax(scores[b,:]) over S=4096. One block (256 thr) per batch.
// ---------------------------------------------------------------------------
__global__ __launch_bounds__(256)
void softmax_kernel(const float* __restrict__ sc, float* __restrict__ attn)
{
    __shared__ float red[256];
    const int b = blockIdx.x, tid = threadIdx.x;
    const float* s = sc + (size_t)b * Sn;

    float vals[16];
    float m = -INFINITY;
#pragma unroll
    for (int j = 0; j < 16; ++j) {
        vals[j] = s[tid + j * 256];
        m = fmaxf(m, vals[j]);
    }
    red[tid] = m; __syncthreads();
    for (int off = 128; off > 0; off >>= 1) {
        if (tid < off) red[tid] = fmaxf(red[tid], red[tid + off]);
        __syncthreads();
    }
    m = red[0]; __syncthreads();

    float sum = 0.f;
#pragma unroll
    for (int j = 0; j < 16; ++j) { vals[j] = __expf(vals[j] - m); sum += vals[j]; }
    red[tid] = sum; __syncthreads();
    for (int off = 128; off > 0; off >>= 1) {
        if (tid < off) red[tid] += red[tid + off];
        __syncthreads();
    }
    const float inv = 1.0f / red[0];
#pragma unroll
    for (int j = 0; j < 16; ++j)
        attn[(size_t)b * Sn + tid + j * 256] = vals[j] * inv;
}

// ---------------------------------------------------------------------------
// part[b,ch,h] = sum_{s in chunk ch} attn[b,s] * enc[b,s,h]
// grid (B, SCH); 256 threads * float4 = all 1024 columns; each s-iteration
// the block issues 256 b128 loads = 4KB contiguous (second 512MB HBM pass).
// attn chunk (2KB) staged in LDS. Deterministic partials (no float atomics).
// ---------------------------------------------------------------------------
__global__ __launch_bounds__(256)
void context_partial(const float* __restrict__ enc, const float* __restrict__ attn,
                     float* __restrict__ part)
{
    __shared__ float sat[SLEN];
    const int b = blockIdx.x, ch = blockIdx.y, tid = threadIdx.x;
    const int s0 = ch * SLEN;

    for (int i = tid; i < SLEN; i += 256)
        sat[i] = attn[(size_t)b * Sn + s0 + i];
    __syncthreads();

    const float4* e4 =
        reinterpret_cast<const float4*>(enc + ((size_t)b * Sn + s0) * Hn) + tid;
    float4 acc = {0.f, 0.f, 0.f, 0.f};
#pragma unroll 4
    for (int s = 0; s < SLEN; ++s) {
        const float a = sat[s];
        const float4 e = e4[(size_t)s * (Hn / 4)];
        acc.x += a * e.x; acc.y += a * e.y;
        acc.z += a * e.z; acc.w += a * e.w;
    }
    reinterpret_cast<float4*>(part + (size_t)(b * SCH + ch) * Hn)[tid] = acc;
}

// ctx[b,h] = sum_ch part[b,ch,h]   (tiny, deterministic reduction)
__global__ __launch_bounds__(256)
void context_reduce(const float* __restrict__ part, float* __restrict__ ctx)
{
    const int b = blockIdx.x, tid = threadIdx.x;
    const float4* p4 = reinterpret_cast<const float4*>(part + (size_t)b * SCH * Hn) + tid;
    float4 acc = {0.f, 0.f, 0.f, 0.f};
#pragma unroll
    for (int c = 0; c < SCH; ++c) {
        const float4 e = p4[(size_t)c * (Hn / 4)];
        acc.x += e.x; acc.y += e.y; acc.z += e.z; acc.w += e.w;
    }
    reinterpret_cast<float4*>(ctx + (size_t)b * Hn)[tid] = acc;
}

// ---------------------------------------------------------------------------
// out[m,n] = tanh( [h1|ctx][m,:] @ C_w[n,:]^T + C_b[n] )   K=2048, WMMA
// ---------------------------------------------------------------------------
__global__ __launch_bounds__(256)
void final_gemm_tanh(const float* __restrict__ h1, const float* __restrict__ ctx,
                     const float* __restrict__ Cw, const float* __restrict__ Cb,
                     float* __restrict__ out)
{
    const int lane = threadIdx.x & 31;
    const int wid  = threadIdx.x >> 5;
    const int half = lane >> 4;
    const int l    = lane & 15;
    const int n0   = (blockIdx.x * 8 + wid) * 16;
    const int m0   = blockIdx.y * 16;

    v8f acc = {0.f, 0.f, 0.f, 0.f, 0.f, 0.f, 0.f, 0.f};
    const float* a1 = h1  + (size_t)(m0 + l) * Hn + half * 2;
    const float* a2 = ctx + (size_t)(m0 + l) * Hn + half * 2;
    const float* bp = Cw  + (size_t)(n0 + l) * (2 * Hn) + half * 2;

    for (int k = 0; k < Hn; k += 4) {
        v2f a, b;
        a.x = a1[k]; a.y = a1[k + 1];
        b.x = bp[k]; b.y = bp[k + 1];
        acc = __builtin_amdgcn_wmma_f32_16x16x4_f32(false, a, false, b,
                                                    (short)0, acc, false, false);
    }
    for (int k = 0; k < Hn; k += 4) {
        v2f a, b;
        a.x = a2[k];      a.y = a2[k + 1];
        b.x = bp[Hn + k]; b.y = bp[Hn + k + 1];
        acc = __builtin_amdgcn_wmma_f32_16x16x4_f32(false, a, false, b,
                                                    (short)0, acc, false, false);
    }
    const float bv = Cb[n0 + l];
#pragma unroll
    for (int r = 0; r < 8; ++r)
        out[(size_t)(m0 + half * 8 + r) * Hn + n0 + l] = tanhf(acc[r] + bv);
}

// ---------------------------------------------------------------------------
extern "C" void kernel_launch(void* const* d_in, const int* in_sizes, int n_in,
                              void* d_out, int out_size, void* d_ws, size_t ws_size,
                              hipStream_t stream)
{
    const float* hidden = (const float*)d_in[0];   // [B,1,H]
    const float* enc    = (const float*)d_in[1];   // [B,S,H]
    const float* A_w    = (const float*)d_in[2];   // [H,H]
    const float* A_b    = (const float*)d_in[3];   // [H]
    const float* C_w    = (const float*)d_in[4];   // [H,2H]
    const float* C_b    = (const float*)d_in[5];   // [H]

    float* out  = (float*)d_out;                   // [B,H]   first 32768 floats
    float* attn = (float*)d_out + Bn * Hn;         // [B,S,1] next 131072 floats

    char*  ws     = (char*)d_ws;
    float* h1     = (float*)(ws);                  // 128 KB
    float* h2     = (float*)(ws + (128 << 10));    // 128 KB
    float* scores = (float*)(ws + (256 << 10));    // 512 KB
    float* ctx    = (float*)(ws + (768 << 10));    // 128 KB
    float* part   = (float*)(ws + (1 << 20));      // 1 MB  [B, SCH, H]

    dim3 gGemm(Hn / 16 / 8, Bn / 16);              // (8, 2): 64 N-tiles x 2 M-tiles

    gemm_xwT_bias<<<gGemm, 256, 0, stream>>>(hidden, A_w, A_b, h1, Hn);
    gemm_xwT_bias<<<gGemm, 256, 0, stream>>>(h1,     A_w, A_b, h2, Hn);
    scores_kernel<<<(Bn * Sn) / 8, 256, 0, stream>>>(enc, h2, scores);
    softmax_kernel<<<Bn, 256, 0, stream>>>(scores, attn);
    context_partial<<<dim3(Bn, SCH), 256, 0, stream>>>(enc, attn, part);
    context_reduce<<<Bn, 256, 0, stream>>>(part, ctx);
    final_gemm_tanh<<<gGemm, 256, 0, stream>>>(h1, ctx, C_w, C_b, out);
}